// Model_51780125721203
// MI455X (gfx1250) — compile-verified
//
#include <hip/hip_runtime.h>
#include <hip/hip_bf16.h>
#include <cstdint>

typedef unsigned int u32;
typedef u32 v4u __attribute__((ext_vector_type(4)));
typedef int  v8i __attribute__((ext_vector_type(8)));
typedef int  v4i __attribute__((ext_vector_type(4)));

static constexpr int S_STATES = 1024;          // 4^5 states
static constexpr int NZ       = 5;             // stay + 4 moves
static constexpr int CCH      = S_STATES * NZ; // 5120 floats per (t, n)
static constexpr int NBUF     = 4;             // score tile ring: up to 3 DMAs in flight

// Issue one TDM load of `nelem` contiguous f32 from global `gptr` into LDS byte
// offset `lds_off`. 1-D tile: tile_dim0 = tensor_dim0 = nelem, 4-byte elements.
// Verified lowering: tensor_load_to_lds s[g0:g0+3], s[g1:g1+7] (groups 2/3 NULL).
__device__ __forceinline__ void tdm_load_row(u32 lds_off, const float* gptr, u32 nelem) {
    uint64_t ga = (uint64_t)(uintptr_t)gptr;
    v4u g0;
    g0.x = 1u;                                        // count=1, user descriptor
    g0.y = lds_off;                                   // lds_addr (bytes)
    g0.z = (u32)(ga & 0xFFFFFFFFu);                   // global_addr[31:0]
    g0.w = (u32)((ga >> 32) & 0x01FFFFFFu)            // global_addr[56:32]
         | (2u << 30);                                // type = 2 ("image")
    v8i g1;
    g1[0] = (int)(2u << 16);                          // data_size = 2 -> 4 bytes
    g1[1] = (int)((nelem & 0xFFFFu) << 16);           // tensor_dim0[15:0]
    g1[2] = (int)((nelem >> 16) | (1u << 16));        // tensor_dim0[31:16], tensor_dim1 = 1
    g1[3] = (int)((nelem & 0xFFFFu) << 16);           // tile_dim0
    g1[4] = 0;                                        // tile_dim1 = tile_dim2 = 0 (unused)
    g1[5] = (int)nelem;                               // tensor_dim0_stride[31:0]
    g1[6] = 0;
    g1[7] = 0;
    v4i z4; z4[0] = 0; z4[1] = 0; z4[2] = 0; z4[3] = 0;   // groups 2/3 unused (<=2D)
    v8i z8; z8[0] = 0; z8[1] = 0; z8[2] = 0; z8[3] = 0;
            z8[4] = 0; z8[5] = 0; z8[6] = 0; z8[7] = 0;
    __builtin_amdgcn_tensor_load_to_lds(g0, g1, z4, z4, z8, 0);
}

__global__ __launch_bounds__(1024, 1)
void ctc_crf_logz_kernel(const float* __restrict__ scores,
                         float* __restrict__ out, int T, int N) {
    __shared__ float alpha[2][S_STATES];     // ping-pong recurrence state (8 KB)
    __shared__ float sbuf[NBUF][CCH];        // score tile ring (4 x 20 KB)
    __shared__ float red[S_STATES];          // final-reduction scratch (4 KB)

    const int s = threadIdx.x;               // one thread per state
    const int n = blockIdx.x;                // one workgroup per batch element

    alpha[0][s] = 0.0f;                      // semiring one = 0.0

    const float* base = scores + (size_t)n * CCH;
    const size_t trow = (size_t)N * CCH;     // float stride between timesteps

    u32 ldsb[NBUF];
#pragma unroll
    for (int b = 0; b < NBUF; ++b) ldsb[b] = (u32)(uintptr_t)&sbuf[b][0];

    // Prologue: fill the pipeline with steps 0..NBUF-2 (buffer NBUF-1 stays free
    // so iteration 0 can issue step NBUF-1 into it).
    if (s == 0) {
#pragma unroll
        for (int k = 0; k < NBUF - 1; ++k) {
            if (k < T) tdm_load_row(ldsb[k], base + (size_t)k * trow, CCH);
        }
    }
    __builtin_amdgcn_s_wait_tensorcnt(NBUF - 2);   // step 0's tile has landed
    __syncthreads();

    int cur = 0;
    const int p = s >> 2;                    // move-predecessor base
    for (int t = 0; t < T; ++t) {
        // Issue step t+NBUF-1 into buf[(t-1)%NBUF], freed by the barrier that
        // ended iteration t-1. Keeps up to NBUF-1 DMAs in flight over compute.
        const int k = t + NBUF - 1;
        if (s == 0 && k < T) {
            tdm_load_row(ldsb[k & (NBUF - 1)], base + (size_t)k * trow, CCH);
        }

        const float* __restrict__ Ms = sbuf[t & (NBUF - 1)];
        const float* __restrict__ a  = alpha[cur];

        const float a0 = a[s];
        const float a1 = a[p];
        const float a2 = a[p + 256];
        const float a3 = a[p + 512];
        const float a4 = a[p + 768];

        const float* m5 = Ms + s * NZ;       // stride-5: gcd(5,64)=1 -> conflict-free
        const float v0 = m5[0] + a0;         // stay
        const float v1 = m5[1] + a1;         // 4 move transitions
        const float v2 = m5[2] + a2;
        const float v3 = m5[3] + a3;
        const float v4 = m5[4] + a4;

        const float mx = fmaxf(fmaxf(fmaxf(v0, v1), fmaxf(v2, v3)), v4);
        const float sum = __expf(v0 - mx) + __expf(v1 - mx) + __expf(v2 - mx)
                        + __expf(v3 - mx) + __expf(v4 - mx);
        alpha[cur ^ 1][s] = mx + __logf(sum);

        // Loads complete in order: tensorcnt <= NBUF-2 guarantees step t+1's
        // tile is resident while t+2 / t+3 stay in flight. Conservative 0 on
        // the short tail. SOPP is uniform; non-issuing waves pass instantly.
        if (t + NBUF <= T) {
            __builtin_amdgcn_s_wait_tensorcnt(NBUF - 2);
        } else {
            __builtin_amdgcn_s_wait_tensorcnt(0);
        }
        __syncthreads();
        cur ^= 1;
    }

    // logZ[n] = logsumexp over states of alpha_T (vT == 0).
    const float av = alpha[cur][s];
    red[s] = av;
    __syncthreads();
    for (int off = 512; off > 0; off >>= 1) {
        if (s < off) red[s] = fmaxf(red[s], red[s + off]);
        __syncthreads();
    }
    const float mx = red[0];
    __syncthreads();
    red[s] = __expf(av - mx);
    __syncthreads();
    for (int off = 512; off > 0; off >>= 1) {
        if (s < off) red[s] += red[s + off];
        __syncthreads();
    }
    if (s == 0) out[n] = mx + __logf(red[0]);
}

extern "C" void kernel_launch(void* const* d_in, const int* in_sizes, int n_in,
                              void* d_out, int out_size, void* d_ws, size_t ws_size,
                              hipStream_t stream) {
    const float* scores = (const float*)d_in[0];
    float* out = (float*)d_out;

    const int N = out_size;                                   // 32 batch elements
    const long long total = (long long)in_sizes[0];           // T*N*C
    const int T = (int)(total / ((long long)N * CCH));        // 2000

    hipLaunchKernelGGL(ctc_crf_logz_kernel, dim3(N), dim3(S_STATES), 0, stream,
                       scores, out, T, N);
}